// OHEMCrossEntropyLoss_40218073759763
// MI455X (gfx1250) — compile-verified
//
#include <hip/hip_runtime.h>
#include <hip/hip_bf16.h>

// OHEM cross-entropy, MI455X (gfx1250, wave32).
// Memory-bound: ~335MB traffic -> ~14us floor @ 23.3 TB/s.
// Wave reduction uses V_WMMA_F32_16X16X4_F32 (layout-independent all-ones-B trick).

typedef __attribute__((ext_vector_type(2))) float v2f;
typedef __attribute__((ext_vector_type(8))) float v8f;

#define OHEM_C 19
#define OHEM_HW (512 * 1024)
#define OHEM_CHW (OHEM_C * OHEM_HW)
#define OHEM_THRESH 0.7f
#define OHEM_IGNORE 255
#define HIST_BINS 256
#define HIST_SCALE 16.0f   // bin = loss * 16, range [0, 16)

// ws layout (dwords): [0]=maskedSum(f32) [1]=maskedCnt(u32)
//                     [2..258)=histCnt(u32)  [258..514)=histSum(f32)
#define WS_DWORDS (2 + HIST_BINS + HIST_BINS)

#if __has_builtin(__builtin_amdgcn_wmma_f32_16x16x4_f32)
#define HAVE_WMMA_F32X4 1
#else
#define HAVE_WMMA_F32X4 0
#endif

// Exact f32 wave32 sum. WMMA path: A(16x4 f32) holds the 32 lane-partials
// (v in a[0], 0 in a[1]); B = all-ones so D[m][n] = rowsum_m(A) independent of
// B's lane layout. Lane-local sum of the 8 D VGPRs covers half the M rows;
// xor-16 shuffle adds the other half -> total = sum over all 64 A slots.
__device__ __forceinline__ float waveReduceSum(float v) {
#if HAVE_WMMA_F32X4
    v2f a; a[0] = v;    a[1] = 0.0f;
    v2f b; b[0] = 1.0f; b[1] = 1.0f;
    v8f c = {};
    v8f d = __builtin_amdgcn_wmma_f32_16x16x4_f32(
        /*neg_a=*/false, a, /*neg_b=*/false, b,
        /*c_mod=*/(short)0, c, /*reuse_a=*/false, /*reuse_b=*/false);
    float r = ((d[0] + d[1]) + (d[2] + d[3])) + ((d[4] + d[5]) + (d[6] + d[7]));
    r += __shfl_xor(r, 16, 32);
    return r;
#else
    for (int off = 16; off > 0; off >>= 1) v += __shfl_xor(v, off, 32);
    return v;
#endif
}

__global__ void ohem_zero_kernel(unsigned* ws, int n) {
    int i = blockIdx.x * blockDim.x + threadIdx.x;
    if (i < n) ws[i] = 0u;
}

__global__ __launch_bounds__(256)
void ohem_main_kernel(const float* __restrict__ logits,
                      const int* __restrict__ targets,
                      unsigned* __restrict__ ws, int npix) {
    float*    gSum = (float*)&ws[0];
    unsigned* gCnt = &ws[1];
    unsigned* gHC  = &ws[2];
    float*    gHS  = (float*)&ws[2 + HIST_BINS];

    __shared__ unsigned sHC[HIST_BINS];
    __shared__ float    sHS[HIST_BINS];
    __shared__ float    sWS[8];
    __shared__ float    sWK[8];

    for (int i = threadIdx.x; i < HIST_BINS; i += blockDim.x) {
        sHC[i] = 0u; sHS[i] = 0.0f;
    }
    __syncthreads();

    float s = 0.0f;  // masked loss sum (this thread)
    float k = 0.0f;  // masked count (this thread)

    const int stride = gridDim.x * blockDim.x;
    for (int p = blockIdx.x * blockDim.x + threadIdx.x; p < npix; p += stride) {
        const int b  = p >> 19;            // / (512*1024)
        const int hw = p & (OHEM_HW - 1);
        const float* base = logits + (size_t)b * OHEM_CHW + hw;

        const int t = targets[p];
        const bool valid = (t != OHEM_IGNORE);
        int tc = t; if (tc < 0) tc = 0; if (tc > OHEM_C - 1) tc = OHEM_C - 1;

        // One pass over memory; all 19 channel values live in registers.
        float x[OHEM_C];
        float xt = 0.0f;
#pragma unroll
        for (int c = 0; c < OHEM_C; ++c) {
            float v = base[(size_t)c * OHEM_HW];
            x[c] = v;
            if (c == tc) xt = v;   // select chain, no dynamic indexing
        }
        float m = x[0];
#pragma unroll
        for (int c = 1; c < OHEM_C; ++c) m = fmaxf(m, x[c]);
        float se = 0.0f;
#pragma unroll
        for (int c = 0; c < OHEM_C; ++c) se += __expf(x[c] - m);

        const float nll  = m + __logf(se) - xt;   // -log_softmax[target]
        const float loss = valid ? nll : 0.0f;

        if (loss > OHEM_THRESH) { s += loss; k += 1.0f; }

        int bin = (int)(loss * HIST_SCALE);
        bin = bin < 0 ? 0 : (bin > HIST_BINS - 1 ? HIST_BINS - 1 : bin);
        atomicAdd(&sHC[bin], 1u);
        atomicAdd(&sHS[bin], loss);
    }

    // Reconverged here (EXEC all-ones): WMMA wave reduction.
    const float ws_s = waveReduceSum(s);
    const float ws_k = waveReduceSum(k);

    const int lane = threadIdx.x & 31;
    const int wave = threadIdx.x >> 5;
    if (lane == 0) { sWS[wave] = ws_s; sWK[wave] = ws_k; }
    __syncthreads();

    if (threadIdx.x == 0) {
        float bs = 0.0f, bk = 0.0f;
        const int nw = blockDim.x >> 5;
        for (int w = 0; w < nw; ++w) { bs += sWS[w]; bk += sWK[w]; }
        atomicAdd(gSum, bs);
        atomicAdd(gCnt, (unsigned)(bk + 0.5f));   // exact: block count <= 4096
    }
    __syncthreads();

    for (int i = threadIdx.x; i < HIST_BINS; i += blockDim.x) {
        atomicAdd(&gHC[i], sHC[i]);
        atomicAdd(&gHS[i], sHS[i]);
    }
}

__global__ void ohem_finalize_kernel(const unsigned* __restrict__ ws,
                                     float* __restrict__ out, int k0) {
    if (blockIdx.x != 0 || threadIdx.x != 0) return;

    const float*    gSum = (const float*)&ws[0];
    const unsigned* gCnt = &ws[1];
    const unsigned* gHC  = &ws[2];
    const float*    gHS  = (const float*)&ws[2 + HIST_BINS];

    const unsigned cnt = *gCnt;
    if (cnt > 0u) {
        out[0] = gSum[0] / (float)cnt;
        return;
    }
    // Fallback: approximate mean of top-k0 losses from 256-bin histogram,
    // scanning from the top bin; partial bin taken pro-rata.
    float s = 0.0f;
    long long got = 0;
    const long long need = (long long)k0;
    for (int b = HIST_BINS - 1; b >= 0 && got < need; --b) {
        const unsigned hc = gHC[b];
        if (hc == 0u) continue;
        const float hs = gHS[b];
        if (got + (long long)hc <= need) {
            s += hs; got += (long long)hc;
        } else {
            const long long take = need - got;
            s += hs * ((float)take / (float)hc);
            got = need;
        }
    }
    out[0] = s / (float)(got > 0 ? got : 1);
}

extern "C" void kernel_launch(void* const* d_in, const int* in_sizes, int n_in,
                              void* d_out, int out_size, void* d_ws, size_t ws_size,
                              hipStream_t stream) {
    const float* logits  = (const float*)d_in[0];
    const int*   targets = (const int*)d_in[1];
    float*       out     = (float*)d_out;
    unsigned*    wsu     = (unsigned*)d_ws;

    const int npix = in_sizes[1];              // B*H*W = 4194304
    int k0 = npix / 10; if (k0 < 1) k0 = 1;    // max(1, int(0.1*N))

    ohem_zero_kernel<<<(WS_DWORDS + 255) / 256, 256, 0, stream>>>(wsu, WS_DWORDS);
    ohem_main_kernel<<<1024, 256, 0, stream>>>(logits, targets, wsu, npix);
    ohem_finalize_kernel<<<1, 32, 0, stream>>>(wsu, out, k0);
}